// DLRM_66331474919974
// MI455X (gfx1250) — compile-verified
//
#include <hip/hip_runtime.h>
#include <math.h>

// ---------------------------------------------------------------------------
// Fused DLRM inference kernel for gfx1250 (MI455X).
// Roofline: ~160MB of embedding gathers dominate (~7us at 23.3 TB/s HBM);
// all matrix math (proj 64x64, top MLP 79->128->64) runs on-chip via fp32
// WMMA (v_wmma_f32_16x16x4_f32) so precision matches the fp32 reference.
// Gather loops are branch-free (clamped index + 0/1 mask) so the compiler
// can pipeline b64 loads instead of serializing on per-iteration branches.
// ---------------------------------------------------------------------------

typedef float v2f __attribute__((ext_vector_type(2)));
typedef float v8f __attribute__((ext_vector_type(8)));

#define DLRM_D     64
#define DLRM_LC    8
#define DLRM_LT    64
#define DLRM_TILE  64      // samples per block
#define DLRM_T1    128
#define DLRM_T2    64
#define DLRM_XDIM  79      // 15 interactions + 64 dense
#define DLRM_XPAD  81      // LDS row stride for x  (81*4B: 17 mod 64 banks, conflict-free)
#define DLRM_H1PAD 132     // LDS row stride for h1 (132*4B: 4 mod 64 banks, conflict-free)
#define DLRM_H2PAD 65      // LDS row stride for h2 (fits inside the x buffer)

__global__ __launch_bounds__(256) void dlrm_fused_kernel(
    const int*   __restrict__ user,      const int*   __restrict__ item,
    const int*   __restrict__ brand,     const int*   __restrict__ cat_idx,
    const int*   __restrict__ cat_len,   const int*   __restrict__ text_idx,
    const int*   __restrict__ text_len,  const float* __restrict__ sales_rank,
    const float* __restrict__ user_tab,  const float* __restrict__ item_tab,
    const float* __restrict__ brand_tab, const float* __restrict__ cat_tab,
    const float* __restrict__ text_tab,
    const float* __restrict__ W_bot,     const float* __restrict__ b_bot,
    const float* __restrict__ W_proj,    const float* __restrict__ b_proj,
    const float* __restrict__ W_t1,      const float* __restrict__ b_t1,
    const float* __restrict__ W_t2,      const float* __restrict__ b_t2,
    const float* __restrict__ W_t3,      const float* __restrict__ b_t3,
    float* __restrict__ out)
{
    // 54.5 KB static LDS. sX holds x (stride 81), later reused as h2 (stride 65).
    // sH1 temporarily holds dense_use (cols 0..63), later layer-1 activations.
    __shared__ float sX [DLRM_TILE * DLRM_XPAD];
    __shared__ float sH1[DLRM_TILE * DLRM_H1PAD];

    const int lane      = threadIdx.x & 31;
    const int wave      = threadIdx.x >> 5;
    const int blockBase = blockIdx.x * DLRM_TILE;
    const int m         = lane & 15;
    const int half      = lane >> 4;

    // =====================================================================
    // Phase A: bottom MLP dense = relu(sr*W_bot + b_bot) -> sX cols 15..78.
    // Lane holds dims (2*lane, 2*lane+1).
    // =====================================================================
    {
        const float2 wb = *(const float2*)&W_bot[2 * lane];
        const float2 bb = *(const float2*)&b_bot[2 * lane];
        for (int i = 0; i < 8; ++i) {
            const int sl = wave * 8 + i;
            const float sr = sales_rank[blockBase + sl];
            float* xrow = &sX[sl * DLRM_XPAD];
            xrow[15 + 2 * lane] = fmaxf(sr * wb.x + bb.x, 0.f);
            xrow[16 + 2 * lane] = fmaxf(sr * wb.y + bb.y, 0.f);
            if (lane == 0) { xrow[79] = 0.f; xrow[80] = 0.f; }   // K-pad for layer 1
        }
    }
    __syncthreads();

    // =====================================================================
    // Phase B: dense_use = dense @ W_proj + b_proj via fp32 WMMA.
    // 16 jobs = 4 sample-tiles x 4 n-tiles, K = 64 (16 steps). A-matrix is
    // the dense block at sX col offset 15; result -> sH1 cols 0..63.
    // =====================================================================
    for (int job = wave; job < 16; job += 8) {
        const int st = job >> 2;
        const int nt = job & 3;
        const int n  = nt * 16 + m;
        const float* xb = &sX[(st * 16 + m) * DLRM_XPAD + 15];
        v8f acc = {0.f, 0.f, 0.f, 0.f, 0.f, 0.f, 0.f, 0.f};
        for (int k0 = 0; k0 < DLRM_D; k0 += 4) {
            const int ka = k0 + half * 2;
            v2f A, B;
            A.x = xb[ka];
            A.y = xb[ka + 1];
            B.x = W_proj[ ka      * DLRM_D + n];
            B.y = W_proj[(ka + 1) * DLRM_D + n];
            acc = __builtin_amdgcn_wmma_f32_16x16x4_f32(
                      false, A, false, B, (short)0, acc, false, false);
        }
        const float bias = b_proj[n];
        #pragma unroll
        for (int r = 0; r < 8; ++r) {
            const int row = half * 8 + r;
            sH1[(st * 16 + row) * DLRM_H1PAD + n] = acc[r] + bias;   // no relu
        }
    }
    __syncthreads();

    // =====================================================================
    // Phase C: embedding gathers + interactions. One wave per sample; lane
    // holds dims (2*lane, 2*lane+1) -> one global_load_b64 per 256B row.
    // Bag loops are branch-free: clamped index (dup loads hit L0) + 0/1 mask.
    // =====================================================================
    for (int i = 0; i < 8; ++i) {
        const int sl = wave * 8 + i;
        const int s  = blockBase + sl;

        float f[6][2];
        {
            const int ui = user[s], ii = item[s], bi = brand[s];
            const float2 ru = *(const float2*)&user_tab [ui * DLRM_D + 2 * lane];
            const float2 ri = *(const float2*)&item_tab [ii * DLRM_D + 2 * lane];
            const float2 rb = *(const float2*)&brand_tab[bi * DLRM_D + 2 * lane];
            f[0][0] = ru.x; f[0][1] = ru.y;
            f[1][0] = ri.x; f[1][1] = ri.y;
            f[2][0] = rb.x; f[2][1] = rb.y;
        }
        // category mean-bag (len may be 0 -> zeros)
        {
            const int clen = cat_len[s];
            const int clc  = (clen > 0 ? clen : 1) - 1;
            float c0 = 0.f, c1 = 0.f;
            #pragma unroll
            for (int j = 0; j < DLRM_LC; ++j) {
                const int jj = (j < clc) ? j : clc;
                const int ci = cat_idx[s * DLRM_LC + jj];
                const float2 row = *(const float2*)&cat_tab[ci * DLRM_D + 2 * lane];
                const float w = (j < clen) ? 1.f : 0.f;
                c0 += w * row.x;  c1 += w * row.y;
            }
            const float cinv = 1.0f / (float)(clen > 0 ? clen : 1);
            f[3][0] = c0 * cinv;  f[3][1] = c1 * cinv;
        }
        // text mean-bag: indices preloaded into 2 regs, broadcast by shuffle
        {
            const int tlen = text_len[s];                 // >= 1 by construction
            const int tlc  = (tlen > 0 ? tlen : 1) - 1;
            const int j0   = (lane      < tlc) ? lane      : tlc;
            const int j1   = (lane + 32 < tlc) ? lane + 32 : tlc;
            const int tj0  = text_idx[s * DLRM_LT + j0];
            const int tj1  = text_idx[s * DLRM_LT + j1];
            float t0 = 0.f, t1 = 0.f;
            #pragma unroll 16
            for (int j = 0; j < DLRM_LT; ++j) {
                const int ti = __shfl((j < 32) ? tj0 : tj1, j & 31, 32);
                const float2 row = *(const float2*)&text_tab[ti * DLRM_D + 2 * lane];
                const float w = (j < tlen) ? 1.f : 0.f;
                t0 += w * row.x;  t1 += w * row.y;
            }
            const float tinv = 1.0f / (float)(tlen > 0 ? tlen : 1);
            f[4][0] = t0 * tinv;  f[4][1] = t1 * tinv;
        }
        // dense_use computed in Phase B
        {
            const float2 du = *(const float2*)&sH1[sl * DLRM_H1PAD + 2 * lane];
            f[5][0] = du.x;  f[5][1] = du.y;
        }
        // 15 upper-triangular interactions: wave-wide dot products over D=64
        float inter[15];
        int p = 0;
        #pragma unroll
        for (int a = 0; a < 6; ++a) {
            #pragma unroll
            for (int b = a + 1; b < 6; ++b) {
                float v = f[a][0] * f[b][0] + f[a][1] * f[b][1];
                v += __shfl_xor(v, 16, 32);
                v += __shfl_xor(v,  8, 32);
                v += __shfl_xor(v,  4, 32);
                v += __shfl_xor(v,  2, 32);
                v += __shfl_xor(v,  1, 32);
                inter[p++] = v;
            }
        }
        if (lane == 0) {
            float* xrow = &sX[sl * DLRM_XPAD];
            #pragma unroll
            for (int q = 0; q < 15; ++q) xrow[q] = inter[q];
        }
    }
    __syncthreads();

    // =====================================================================
    // Layer 1: h1 = relu(x @ W_t1 + b_t1), 64x79 @ 79x128 via fp32 WMMA.
    // 32 jobs = 4 sample-tiles x 8 n-tiles; K padded 79 -> 80 (20 steps).
    // Overwrites sH1 (dense_use already consumed).
    // =====================================================================
    for (int job = wave; job < 32; job += 8) {
        const int st = job >> 3;
        const int nt = job & 7;
        const int n  = nt * 16 + m;
        const float* xb = &sX[(st * 16 + m) * DLRM_XPAD];
        v8f acc = {0.f, 0.f, 0.f, 0.f, 0.f, 0.f, 0.f, 0.f};
        for (int k0 = 0; k0 < 80; k0 += 4) {
            const int ka = k0 + half * 2;
            v2f A, B;
            A.x = xb[ka];
            A.y = xb[ka + 1];
            B.x = (ka     < DLRM_XDIM) ? W_t1[ ka      * DLRM_T1 + n] : 0.f;
            B.y = (ka + 1 < DLRM_XDIM) ? W_t1[(ka + 1) * DLRM_T1 + n] : 0.f;
            acc = __builtin_amdgcn_wmma_f32_16x16x4_f32(
                      false, A, false, B, (short)0, acc, false, false);
        }
        const float bias = b_t1[n];
        #pragma unroll
        for (int r = 0; r < 8; ++r) {
            const int row = half * 8 + r;
            sH1[(st * 16 + row) * DLRM_H1PAD + n] = fmaxf(acc[r] + bias, 0.f);
        }
    }
    __syncthreads();

    // =====================================================================
    // Layer 2: h2 = relu(h1 @ W_t2 + b_t2), 64x128 @ 128x64 via fp32 WMMA.
    // 16 jobs; K = 128 (32 steps). h2 written into the retired x buffer.
    // =====================================================================
    for (int job = wave; job < 16; job += 8) {
        const int st = job >> 2;
        const int nt = job & 3;
        const int n  = nt * 16 + m;
        const float* hb = &sH1[(st * 16 + m) * DLRM_H1PAD];
        v8f acc = {0.f, 0.f, 0.f, 0.f, 0.f, 0.f, 0.f, 0.f};
        for (int k0 = 0; k0 < DLRM_T1; k0 += 4) {
            const int ka = k0 + half * 2;
            v2f A, B;
            A.x = hb[ka];
            A.y = hb[ka + 1];
            B.x = W_t2[ ka      * DLRM_T2 + n];
            B.y = W_t2[(ka + 1) * DLRM_T2 + n];
            acc = __builtin_amdgcn_wmma_f32_16x16x4_f32(
                      false, A, false, B, (short)0, acc, false, false);
        }
        const float bias = b_t2[n];
        #pragma unroll
        for (int r = 0; r < 8; ++r) {
            const int row = half * 8 + r;
            sX[(st * 16 + row) * DLRM_H2PAD + n] = fmaxf(acc[r] + bias, 0.f);
        }
    }
    __syncthreads();

    // =====================================================================
    // Layer 3 + sigmoid: logit = h2 @ W_t3 + b_t3. 4 threads per sample.
    // =====================================================================
    {
        const int sl = threadIdx.x >> 2;   // 0..63
        const int q  = threadIdx.x & 3;
        float psum = 0.f;
        #pragma unroll
        for (int j = 0; j < 16; ++j) {
            const int n = q * 16 + j;
            psum += sX[sl * DLRM_H2PAD + n] * W_t3[n];
        }
        psum += __shfl_xor(psum, 1, 32);
        psum += __shfl_xor(psum, 2, 32);
        if (q == 0) {
            const float logit = psum + b_t3[0];
            out[blockBase + sl] = 1.0f / (1.0f + __expf(-logit));
        }
    }
}

extern "C" void kernel_launch(void* const* d_in, const int* in_sizes, int n_in,
                              void* d_out, int out_size, void* d_ws, size_t ws_size,
                              hipStream_t stream) {
    (void)n_in; (void)d_ws; (void)ws_size; (void)out_size;
    const int*   user       = (const int*)  d_in[0];
    const int*   item       = (const int*)  d_in[1];
    const int*   brand      = (const int*)  d_in[2];
    const int*   cat_idx    = (const int*)  d_in[3];
    const int*   cat_len    = (const int*)  d_in[4];
    const int*   text_idx   = (const int*)  d_in[5];
    const int*   text_len   = (const int*)  d_in[6];
    const float* sales_rank = (const float*)d_in[7];
    const float* user_tab   = (const float*)d_in[8];
    const float* item_tab   = (const float*)d_in[9];
    const float* brand_tab  = (const float*)d_in[10];
    const float* cat_tab    = (const float*)d_in[11];
    const float* text_tab   = (const float*)d_in[12];
    const float* W_bot      = (const float*)d_in[13];
    const float* b_bot      = (const float*)d_in[14];
    const float* W_proj     = (const float*)d_in[15];
    const float* b_proj     = (const float*)d_in[16];
    const float* W_t1       = (const float*)d_in[17];
    const float* b_t1       = (const float*)d_in[18];
    const float* W_t2       = (const float*)d_in[19];
    const float* b_t2       = (const float*)d_in[20];
    const float* W_t3       = (const float*)d_in[21];
    const float* b_t3       = (const float*)d_in[22];
    float* out = (float*)d_out;

    const int batch  = in_sizes[0];
    const int blocks = batch / DLRM_TILE;   // 16384 / 64 = 256

    dlrm_fused_kernel<<<blocks, 256, 0, stream>>>(
        user, item, brand, cat_idx, cat_len, text_idx, text_len, sales_rank,
        user_tab, item_tab, brand_tab, cat_tab, text_tab,
        W_bot, b_bot, W_proj, b_proj, W_t1, b_t1, W_t2, b_t2, W_t3, b_t3,
        out);
}